// CNN4D_60610578481421
// MI455X (gfx1250) — compile-verified
//
#include <hip/hip_runtime.h>
#include <stdint.h>

#define LAYERS 4
#define BB 2
#define CC 3
#define DD1 48
#define DD2 48
#define DD3 48
#define DD4 24

// padded dims (halo = 1 on each side of every spatial dim)
#define PD1 50
#define PD2 50
#define PD3 50
#define PD4 26

static constexpr int NPIX   = BB * DD1 * DD2 * DD3 * DD4;     // 5,308,416
static constexpr int NELEM  = NPIX * CC;                      // 15,925,248
static constexpr int SP     = DD1 * DD2 * DD3 * DD4;          // 2,654,208 (per-batch spatial)
static constexpr int PP4    = PD4;                            // 26
static constexpr int PP34   = PD3 * PD4;                      // 1,300
static constexpr int PP234  = PD2 * PD3 * PD4;                // 65,000
static constexpr int PP1234 = PD1 * PD2 * PD3 * PD4;          // 3,250,000
static constexpr int NP6    = BB * 6 * PP1234;                // 39,000,000
static constexpr int NP3    = BB * 3 * PP1234;                // 19,500,000

typedef __attribute__((ext_vector_type(2))) float    v2f;
typedef __attribute__((ext_vector_type(8))) float    v8f;
typedef __attribute__((ext_vector_type(4))) uint32_t u32x4;
typedef __attribute__((ext_vector_type(8))) uint32_t u32x8;

__device__ __forceinline__ void decomp_pix(int p, int& b, int& d1, int& d2, int& d3, int& d4) {
    d4 = p % DD4; p /= DD4;
    d3 = p % DD3; p /= DD3;
    d2 = p % DD2; p /= DD2;
    d1 = p % DD1; b  = p / DD1;
}

// ---------------------------------------------------------------------------
// Padding / staging kernels (pure bandwidth; padded tensors stay L2-resident)
// ---------------------------------------------------------------------------
__global__ __launch_bounds__(256) void init_p6_kernel(
    const float* __restrict__ bnd, float* __restrict__ p6)
{
    int idx = blockIdx.x * blockDim.x + threadIdx.x;
    if (idx >= NP6) return;
    int t = idx;
    const int p4 = t % PD4; t /= PD4;
    const int p3 = t % PD3; t /= PD3;
    const int p2 = t % PD2; t /= PD2;
    const int p1 = t % PD1; t /= PD1;
    const int ch = t % 6;
    const int b  = t / 6;
    float v = 0.f;
    if (ch >= 3 &&
        (unsigned)(p1 - 1) < (unsigned)DD1 && (unsigned)(p2 - 1) < (unsigned)DD2 &&
        (unsigned)(p3 - 1) < (unsigned)DD3 && (unsigned)(p4 - 1) < (unsigned)DD4) {
        v = bnd[(((((b * CC) + (ch - 3)) * DD1 + (p1 - 1)) * DD2 + (p2 - 1)) * DD3 + (p3 - 1)) * DD4 + (p4 - 1)];
    }
    p6[idx] = v;
}

__global__ __launch_bounds__(256) void init_t1p_kernel(float* __restrict__ t1p)
{
    int idx = blockIdx.x * blockDim.x + threadIdx.x;
    if (idx < NP3) t1p[idx] = 0.f;
}

__global__ __launch_bounds__(256) void padcat_cur_kernel(
    const float* __restrict__ cur, float* __restrict__ p6)
{
    const int idx = blockIdx.x * blockDim.x + threadIdx.x;   // exact grid: NELEM/256
    int t = idx;
    const int d4 = t % DD4; t /= DD4;
    const int d3 = t % DD3; t /= DD3;
    const int d2 = t % DD2; t /= DD2;
    const int d1 = t % DD1; t /= DD1;
    const int c  = t % CC;
    const int b  = t / CC;
    p6[(((((b * 6) + c) * PD1 + (d1 + 1)) * PD2 + (d2 + 1)) * PD3 + (d3 + 1)) * PD4 + (d4 + 1)] = cur[idx];
}

// ---------------------------------------------------------------------------
// Per-layer weight image builder: contiguous [16 x KPAD] zero-padded B matrix
// followed by [KPAD] tap-delta table, ready for one TDM pull into LDS.
// ---------------------------------------------------------------------------
template<int CIN, int KV1, int KV2, int KV3, int KV4>
__global__ __launch_bounds__(256) void build_wimg_kernel(
    const float* __restrict__ W, uint32_t* __restrict__ dst)
{
    constexpr int KVOL = KV1 * KV2 * KV3 * KV4;
    constexpr int KTOT = CIN * KVOL;
    constexpr int KPAD = ((KTOT + 3) / 4) * 4;
    constexpr int TW   = 16 * KPAD + KPAD;
    const int i = blockIdx.x * blockDim.x + threadIdx.x;
    if (i >= TW) return;
    if (i < 16 * KPAD) {
        const int row = i / KPAD, k = i % KPAD;
        const float v = (row < 3 && k < KTOT) ? W[row * KTOT + k] : 0.f;
        dst[i] = __float_as_uint(v);
    } else {
        const int k = i - 16 * KPAD;
        int d = 0;
        if (k < KTOT) {
            const int ci = k / KVOL;
            const int rk = k % KVOL;
            const int o1 = rk / (KV2 * KV3 * KV4);
            const int o2 = (rk / (KV3 * KV4)) % KV2;
            const int o3 = (rk / KV4) % KV3;
            const int o4 = rk % KV4;
            d = (((ci * PD1 + o1) * PD2 + o2) * PD3 + o3) * PD4 + o4;
        }
        dst[i] = (uint32_t)d;
    }
}

// ---------------------------------------------------------------------------
// WMMA conv on a zero-padded CIN-channel tensor. Weight image + delta table
// staged LDS-side by the Tensor Data Mover (TENSORcnt), then the K loop runs
// ds_load_b64 pairs + 2 global gathers + v_wmma_f32_16x16x4_f32 per chunk.
// ---------------------------------------------------------------------------
template<int CIN, int KV1, int KV2, int KV3, int KV4, int S4, bool OUTPAD>
__global__ __launch_bounds__(256) void conv4d_wmma_kernel(
    const float*    __restrict__ in,     // padded (BB, CIN, PD1, PD2, PD3, PD4)
    const uint32_t* __restrict__ wimg,   // [16*KPAD weights | KPAD deltas]
    const float*    __restrict__ bias,   // (3)
    float*          __restrict__ outp)
{
    constexpr int KVOL = KV1 * KV2 * KV3 * KV4;
    constexpr int KTOT = CIN * KVOL;
    constexpr int NCH  = (KTOT + 3) / 4;
    constexpr int KPAD = NCH * 4;
    constexpr int TW   = 16 * KPAD + KPAD;

    __shared__ __align__(16) uint32_t shbuf[TW];     // sole LDS block -> offset 0

    if (threadIdx.x < 32) {
        // Tensor DMA descriptor (CDNA5 ISA), normal mode,
        // one row of TW dwords: global wimg -> LDS offset 0.
        const uint64_t ga = (uint64_t)(uintptr_t)wimg;
        union { u32x4 v; uint32_t s[4]; } g0;
        union { u32x8 v; uint32_t s[8]; } g1;
        g0.s[0] = 1u;                                   // count=1, user descriptor
        g0.s[1] = 0u;                                   // lds_addr = 0
        g0.s[2] = (uint32_t)ga;                         // global_addr[31:0]
        g0.s[3] = (uint32_t)(ga >> 32) | (2u << 30);    // global_addr[56:32] | type=2
        g1.s[0] = (2u << 16);                           // wg_mask=0, data_size=4B
        g1.s[1] = ((uint32_t)TW & 0xFFFFu) << 16;       // tensor_dim0[15:0]
        g1.s[2] = ((uint32_t)TW >> 16) | (1u << 16);    // tensor_dim0[31:16] | tensor_dim1=1
        g1.s[3] = ((uint32_t)TW & 0xFFFFu) << 16;       // tile_dim0 = TW
        g1.s[4] = 0u;                                   // tile_dim1=0 (1D), tile_dim2=0
        g1.s[5] = (uint32_t)TW;                         // tensor_dim0_stride[31:0]
        g1.s[6] = 0u;
        g1.s[7] = 0u;
        asm volatile("tensor_load_to_lds %0, %1\n\t"
                     "s_wait_tensorcnt 0x0"
                     :: "s"(g0.v), "s"(g1.v)
                     : "memory");
    }
    __syncthreads();

    const float* wlds = (const float*)shbuf;
    const int*   dtab = (const int*)(shbuf + 16 * KPAD);

    const int lane = threadIdx.x & 31;
    const int wave = (threadIdx.x >> 5) + (blockIdx.x << 3);
    const int base = wave * 16;
    const int m    = lane & 15;
    const int hi   = lane >> 4;

    int b, d1, d2, d3, d4;
    decomp_pix(base + m, b, d1, d2, d3, d4);
    const int vbase = (b * CIN) * PP1234 + d1 * PP234 + d2 * PP34 + d3 * PP4 + (d4 + S4);
    const int trow  = m * KPAD;

    v8f acc = {0.f, 0.f, 0.f, 0.f, 0.f, 0.f, 0.f, 0.f};
    for (int kc = 0; kc < NCH; ++kc) {
        const int k0 = kc * 4 + hi * 2;                               // even -> 8B aligned
        const int2   dd = *reinterpret_cast<const int2*>(&dtab[k0]);  // ds_load_b64
        const float2 wv = *reinterpret_cast<const float2*>(&wlds[trow + k0]);
        const v2f av = { in[vbase + dd.x], in[vbase + dd.y] };
        const v2f bv = { wv.x, wv.y };
        acc = __builtin_amdgcn_wmma_f32_16x16x4_f32(
                  false, av, false, bv, (short)0, acc, false, false);
    }

    // D[mrow, n] lives in lane (n + 16*(mrow>=8)), VGPR mrow&7; here n = m.
    if (m < 3) {
        const float bia = bias[m];
        const int   q   = base + hi * 8;
        if (OUTPAD) {
            int qb, c1, c2, c3, c4;
            decomp_pix(q, qb, c1, c2, c3, c4);
            int idx = (qb * 3 + m) * PP1234 + (c1 + 1) * PP234 + (c2 + 1) * PP34 +
                      (c3 + 1) * PP4 + (c4 + 1);
#pragma unroll
            for (int j = 0; j < 8; ++j) {
                outp[idx] = acc[j] + bia;
                ++c4; ++idx;                             // advance one pixel, carry chain
                if (c4 == DD4) {
                    c4 = 0; idx += PP4 - DD4; ++c3;
                    if (c3 == DD3) {
                        c3 = 0; idx += PP34 - DD3 * PP4; ++c2;
                        if (c2 == DD2) {
                            c2 = 0; idx += PP234 - DD2 * PP34; ++c1;
                            if (c1 == DD1) { c1 = 0; idx += 3 * PP1234 - DD1 * PP234; }
                        }
                    }
                }
            }
        } else {
            // standard layout is affine in q within a batch: idx = q + m*SP + qb*(CC-1)*SP
#pragma unroll
            for (int j = 0; j < 8; ++j) {
                const int qq  = q + j;
                const int adj = (qq >= SP) ? ((CC - 1) * SP) : 0;   // BB == 2
                outp[qq + m * SP + adj] = acc[j] + bia;
            }
        }
    }
}

// out = conv(t2, W2 (1,1,1,3), pad (0,0,0,1)) + b2 + conv(cur, Wd 1x1x1x1) + bd
__global__ __launch_bounds__(256) void combine_kernel(
    const float* __restrict__ t2, const float* __restrict__ cur,
    const float* __restrict__ W2, const float* __restrict__ b2,
    const float* __restrict__ Wd, const float* __restrict__ bd,
    float* __restrict__ outp)
{
    const int p = blockIdx.x * blockDim.x + threadIdx.x;    // exact grid
    int b, d1, d2, d3, d4;
    decomp_pix(p, b, d1, d2, d3, d4);
    const int sbase = ((((b * CC) * DD1 + d1) * DD2 + d2) * DD3 + d3) * DD4 + d4;

    float tin[3][3], rin[3];
#pragma unroll
    for (int ci = 0; ci < 3; ++ci) {
        rin[ci] = cur[sbase + ci * SP];
#pragma unroll
        for (int k4 = 0; k4 < 3; ++k4) {
            const int c4 = d4 + k4 - 1;
            tin[ci][k4] = ((unsigned)c4 < (unsigned)DD4)
                              ? t2[sbase + ci * SP + (k4 - 1)] : 0.f;
        }
    }
#pragma unroll
    for (int co = 0; co < 3; ++co) {
        float acc = b2[co] + bd[co];
#pragma unroll
        for (int ci = 0; ci < 3; ++ci) {
            acc += rin[ci] * Wd[co * 3 + ci];
#pragma unroll
            for (int k4 = 0; k4 < 3; ++k4)
                acc += tin[ci][k4] * W2[(co * 3 + ci) * 3 + k4];
        }
        outp[sbase + co * SP] = acc;
    }
}

extern "C" void kernel_launch(void* const* d_in, const int* in_sizes, int n_in,
                              void* d_out, int out_size, void* d_ws, size_t ws_size,
                              hipStream_t stream)
{
    const float* f   = (const float*)d_in[0];
    const float* bnd = (const float*)d_in[1];
    const float* Wg  = (const float*)d_in[2];
    const float* bg  = (const float*)d_in[3];
    const float* W1  = (const float*)d_in[4];
    const float* b1  = (const float*)d_in[5];
    const float* W2  = (const float*)d_in[6];
    const float* b2  = (const float*)d_in[7];
    const float* Wd  = (const float*)d_in[8];
    const float* bd  = (const float*)d_in[9];

    constexpr int KPAD_G = ((486 + 3) / 4) * 4;          // 488
    constexpr int KPAD_1 = ((81 + 3) / 4) * 4;           // 84
    constexpr int TW_G   = 16 * KPAD_G + KPAD_G;         // 8,296 dwords
    constexpr int TW_1   = 16 * KPAD_1 + KPAD_1;         // 1,428 dwords

    float* ws   = (float*)d_ws;
    float* p6   = ws;                                    // 39.0M floats
    float* t1p  = p6  + (size_t)NP6;                     // 19.5M floats
    float* t2   = t1p + (size_t)NP3;
    float* bufA = t2   + (size_t)NELEM;
    float* bufB = bufA + (size_t)NELEM;
    uint32_t* wimg_g = (uint32_t*)(bufB + (size_t)NELEM);
    uint32_t* wimg_1 = wimg_g + TW_G;
    float* outs[LAYERS] = { bufA, bufB, bufA, (float*)d_out };

    const dim3 blk(256);
    const dim3 cgrd(NPIX / 16 / 8);                      // 41,472 — exact
    const dim3 egrd(NPIX / 256);                         // 20,736 — exact
    const dim3 pgrd(NELEM / 256);                        // 62,208 — exact

    init_p6_kernel <<<dim3((NP6 + 255) / 256), blk, 0, stream>>>(bnd, p6);
    init_t1p_kernel<<<dim3((NP3 + 255) / 256), blk, 0, stream>>>(t1p);

    const float* cur = f;
    for (int i = 0; i < LAYERS; ++i) {
        padcat_cur_kernel<<<pgrd, blk, 0, stream>>>(cur, p6);
        build_wimg_kernel<6, 3, 3, 3, 3>
            <<<dim3((TW_G + 255) / 256), blk, 0, stream>>>(Wg + (size_t)i * 3 * 486, wimg_g);
        build_wimg_kernel<3, 3, 3, 3, 1>
            <<<dim3((TW_1 + 255) / 256), blk, 0, stream>>>(W1 + (size_t)i * 3 * 81, wimg_1);
        conv4d_wmma_kernel<6, 3, 3, 3, 3, /*S4=*/0, /*OUTPAD=*/true>
            <<<cgrd, blk, 0, stream>>>(p6, wimg_g, bg + i * 3, t1p);
        conv4d_wmma_kernel<3, 3, 3, 3, 1, /*S4=*/1, /*OUTPAD=*/false>
            <<<cgrd, blk, 0, stream>>>(t1p, wimg_1, b1 + i * 3, t2);
        combine_kernel<<<egrd, blk, 0, stream>>>(
            t2, cur, W2 + (size_t)i * 27, b2 + i * 3, Wd + (size_t)i * 9, bd + i * 3, outs[i]);
        cur = outs[i];
    }
}